// GraphConv_1786706395354
// MI455X (gfx1250) — compile-verified
//
#include <hip/hip_runtime.h>
#include <hip/hip_bf16.h>

// GraphConv (norm='both') for N nodes, D=64 fp32 features, E edges.
//
// Roofline: feat (25.6MB) + h_neigh (25.6MB) fit in MI455X's 192MB L2, so the
// 320MB of gathered feature rows and 80M scatter atomics are L2 traffic, not
// HBM. HBM moves only ~90MB total (~4us @ 23.3TB/s). Bottleneck = L2 atomic
// throughput -> wave-per-edge fp32 atomics, float2 per lane (wave32: 32 lanes
// x 8B = one 256B feature row per wave).
//
// Finalize pass out = (feat + h) * diag(norm_r) is expressed as an exact
// 16x16x4 F32 WMMA chain (diag matrix => one nonzero product per dot), which
// lights up the CDNA5 matrix pipe at zero accuracy cost.

#define FD 64  // feature dim

typedef __attribute__((ext_vector_type(2))) float v2f;
typedef __attribute__((ext_vector_type(8))) float v8f;

// ---------------- degree histograms ----------------
__global__ void gc_deg_kernel(const int* __restrict__ src,
                              const int* __restrict__ dst,
                              float* __restrict__ odeg,
                              float* __restrict__ ideg, int E) {
  int i = blockIdx.x * blockDim.x + threadIdx.x;
  int stride = gridDim.x * blockDim.x;
  for (int e = i; e < E; e += stride) {
    atomicAdd(&odeg[src[e]], 1.0f);
    atomicAdd(&ideg[dst[e]], 1.0f);
  }
}

// ---------------- norms ----------------
__global__ void gc_norm_kernel(const float* __restrict__ odeg,
                               const float* __restrict__ ideg,
                               float* __restrict__ nl,
                               float* __restrict__ nr, int N) {
  int i = blockIdx.x * blockDim.x + threadIdx.x;
  if (i < N) {
    nl[i] = rsqrtf(fmaxf(odeg[i], 1.0f) + 1.0f);
    nr[i] = rsqrtf(fmaxf(ideg[i], 1.0f) + 1.0f);
  }
}

// ---------------- edge scatter: h[dst] += feat[src] * nl[src] ----------------
// One wave (32 lanes) per edge: each lane handles a float2 of the 64-wide row.
__global__ void gc_scatter_kernel(const float* __restrict__ feat,
                                  const int* __restrict__ src,
                                  const int* __restrict__ dst,
                                  const float* __restrict__ nl,
                                  float* __restrict__ h, int E) {
  const int wavesPerBlock = blockDim.x >> 5;
  const int wid = blockIdx.x * wavesPerBlock + (threadIdx.x >> 5);
  const int lane = threadIdx.x & 31;
  const int nw = gridDim.x * wavesPerBlock;
  for (int e = wid; e < E; e += nw) {
    const int s = src[e];
    const int d = dst[e];
    const int en = e + nw;
    if (en < E) {
      // gfx1250 global_prefetch_b8: pull next edge's feature row toward us.
      __builtin_prefetch(&feat[(size_t)src[en] * FD + lane * 2], 0, 3);
    }
    const float w = nl[s];
    const float2 v =
        *reinterpret_cast<const float2*>(&feat[(size_t)s * FD + lane * 2]);
    float* hp = &h[(size_t)d * FD + lane * 2];
    atomicAdd(hp, v.x * w);
    atomicAdd(hp + 1, v.y * w);
  }
}

// ---------------- finalize: out = (feat + h) * nr[:,None] ----------------
// WMMA formulation: for each 16-node tile, D(16x16) = diag(nr)(16x16) x
// M(16x16), split over K as four V_WMMA_F32_16X16X4_F32. Exact in fp32:
// every dot product has exactly one nonzero term (nr[m] * M[m][n]).
//
// ISA layouts (wave32):
//   A 16x4 : lane m (both halves) holds row m; VGPR0 = K{0|2}, VGPR1 = K{1|3}
//            (lanes 0-15 -> K0/K1, lanes 16-31 -> K2/K3)
//   B 4x16 : lane n holds column n&15; VGPR0 = K{0|2}, VGPR1 = K{1|3}
//   D 16x16: VGPR r: lanes 0-15 -> row r, lanes 16-31 -> row r+8, col = lane&15
__global__ void __launch_bounds__(256)
gc_finalize_wmma(const float* __restrict__ feat, const float* __restrict__ h,
                 const float* __restrict__ nr, float* __restrict__ out,
                 int ntiles) {
  const int wavesPerBlock = blockDim.x >> 5;
  const int tile = blockIdx.x * wavesPerBlock + (threadIdx.x >> 5);
  if (tile >= ntiles) return;  // uniform per wave: EXEC stays all-ones inside
  const int lane = threadIdx.x & 31;
  const int m = lane & 15;
  const int half = lane >> 4;
  const int nb = tile * 16;          // node base
  const float rl = nr[nb + m];       // diag value for this lane's row

#if __has_builtin(__builtin_amdgcn_wmma_f32_16x16x4_f32)
  for (int ct = 0; ct < 4; ++ct) {   // four 16-col tiles of the 64-wide row
    const int cb = ct * 16;
    v8f c = {};
#pragma unroll
    for (int k = 0; k < 4; ++k) {
      // A_k = columns [4k,4k+4) of diag(nr)
      v2f a;
      a.x = (m == 4 * k + 2 * half) ? rl : 0.0f;
      a.y = (m == 4 * k + 2 * half + 1) ? rl : 0.0f;
      // B_k = rows [4k,4k+4) of M = feat + h
      const int r0 = 4 * k + 2 * half;
      const size_t i0 = (size_t)(nb + r0) * FD + cb + m;
      const size_t i1 = i0 + FD;  // next row
      v2f b;
      b.x = feat[i0] + h[i0];
      b.y = feat[i1] + h[i1];
      c = __builtin_amdgcn_wmma_f32_16x16x4_f32(false, a, false, b, (short)0,
                                                c, false, false);
    }
#pragma unroll
    for (int r = 0; r < 8; ++r) {
      out[(size_t)(nb + r + 8 * half) * FD + cb + m] = c[r];
    }
  }
#else
  // Scalar fallback (same math)
  for (int r2 = half; r2 < 16; r2 += 2) {
    const size_t base = (size_t)(nb + r2) * FD;
    const float w = nr[nb + r2];
    for (int cidx = m; cidx < FD; cidx += 16) {
      out[base + cidx] = (feat[base + cidx] + h[base + cidx]) * w;
    }
  }
#endif
}

// scalar tail for N % 16 nodes (not launched when N % 16 == 0)
__global__ void gc_finalize_tail(const float* __restrict__ feat,
                                 const float* __restrict__ h,
                                 const float* __restrict__ nr,
                                 float* __restrict__ out, int start, int N) {
  int idx = blockIdx.x * blockDim.x + threadIdx.x;
  int total = (N - start) * FD;
  if (idx < total) {
    int node = start + idx / FD;
    int c = idx % FD;
    size_t i = (size_t)node * FD + c;
    out[i] = (feat[i] + h[i]) * nr[node];
  }
}

extern "C" void kernel_launch(void* const* d_in, const int* in_sizes, int n_in,
                              void* d_out, int out_size, void* d_ws,
                              size_t ws_size, hipStream_t stream) {
  const float* feat = (const float*)d_in[0];
  const int* src = (const int*)d_in[1];
  const int* dst = (const int*)d_in[2];
  float* out = (float*)d_out;

  const int N = in_sizes[0] / FD;
  const int E = in_sizes[1];

  // Workspace layout
  float* h = (float*)d_ws;                  // N*FD
  float* odeg = h + (size_t)N * FD;         // N
  float* ideg = odeg + N;                   // N
  float* nl = ideg + N;                     // N
  float* nr = nl + N;                       // N

  // Zero h + odeg + ideg (contiguous)
  hipMemsetAsync(d_ws, 0, ((size_t)N * FD + 2 * (size_t)N) * sizeof(float),
                 stream);

  // 1) degrees
  {
    int blocks = (E + 255) / 256;
    if (blocks > 4096) blocks = 4096;
    gc_deg_kernel<<<blocks, 256, 0, stream>>>(src, dst, odeg, ideg, E);
  }
  // 2) norms
  {
    int blocks = (N + 255) / 256;
    gc_norm_kernel<<<blocks, 256, 0, stream>>>(odeg, ideg, nl, nr, N);
  }
  // 3) edge scatter (wave-per-edge)
  {
    int wavesNeeded = E;  // one edge per wave minimum pass
    int blocks = (wavesNeeded + 7) / 8;
    if (blocks > 8192) blocks = 8192;  // grid-stride covers the rest
    gc_scatter_kernel<<<blocks, 256, 0, stream>>>(feat, src, dst, nl, h, E);
  }
  // 4) finalize (WMMA over full 16-node tiles, scalar tail otherwise)
  {
    const int ntiles = N / 16;
    if (ntiles > 0) {
      int blocks = (ntiles + 7) / 8;  // 8 waves/block, 1 tile/wave
      gc_finalize_wmma<<<blocks, 256, 0, stream>>>(feat, h, nr, out, ntiles);
    }
    const int rem = N - ntiles * 16;
    if (rem > 0) {
      int blocks = (rem * FD + 255) / 256;
      gc_finalize_tail<<<blocks, 256, 0, stream>>>(feat, h, nr, out,
                                                   ntiles * 16, N);
    }
  }
}